// HolographicTTEmbedding_4475355922619
// MI455X (gfx1250) — compile-verified
//
#include <hip/hip_runtime.h>

typedef __attribute__((ext_vector_type(2))) float v2f;
typedef __attribute__((ext_vector_type(8))) float v8f;

#define TT_V2   224   // second factor of vocab decomposition
#define TT_RANK 16
#define TT_D1   32
#define TT_D2   32

// One wave per token: out(32x32) = (cos(phase)*c1)^T (32x16)  x  c2 (16x32)
// Four 16x16 output tiles, K=16 as four k-steps of 4 via V_WMMA_F32_16X16X4_F32.
__global__ __launch_bounds__(256) void tt_embed_wmma_kernel(
    const int*   __restrict__ ids,     // (ntok) token ids
    const float* __restrict__ core1,   // (225, 16, 32)
    const float* __restrict__ core2,   // (224, 16, 32)
    const float* __restrict__ phase,   // (16)
    float*       __restrict__ out,     // (ntok, 1024)
    int ntok)
{
    const int lane  = threadIdx.x & 31;
    const int hi    = lane >> 4;        // 0: lanes 0-15, 1: lanes 16-31
    const int lo    = lane & 15;
    const int wave  = (int)((blockIdx.x * blockDim.x + threadIdx.x) >> 5);
    const int nwav  = (int)((gridDim.x * blockDim.x) >> 5);

    // Precompute cos(phase[r]) for the ranks this lane touches:
    // r = 4*k + 2*hi + j  (k = k-step, j = A/B vgpr index)
    float pm[4][2];
#pragma unroll
    for (int k = 0; k < 4; ++k) {
#pragma unroll
        for (int j = 0; j < 2; ++j)
            pm[k][j] = cosf(phase[4 * k + 2 * hi + j]);
    }

    for (int t = wave; t < ntok; t += nwav) {
        int id = ids[t];
        int i1 = id / TT_V2;                 // 0..224 (fits V1=225, no clip needed)
        int i2 = id - i1 * TT_V2;            // 0..223
        const float* __restrict__ c1 = core1 + (size_t)i1 * (TT_RANK * TT_D1);
        const float* __restrict__ c2 = core2 + (size_t)i2 * (TT_RANK * TT_D2);

        v8f acc00 = {}, acc01 = {}, acc10 = {}, acc11 = {};

#pragma unroll
        for (int k = 0; k < 4; ++k) {
            const int r0 = 4 * k + 2 * hi;   // K rows held by this lane half
            v2f a0, a1, b0, b1;
            // A tiles (M = d1): a0 -> d1 in [0,16), a1 -> d1 in [16,32)
            a0.x = c1[(r0    ) * TT_D1 + lo     ] * pm[k][0];
            a0.y = c1[(r0 + 1) * TT_D1 + lo     ] * pm[k][1];
            a1.x = c1[(r0    ) * TT_D1 + 16 + lo] * pm[k][0];
            a1.y = c1[(r0 + 1) * TT_D1 + 16 + lo] * pm[k][1];
            // B tiles (N = d2): b0 -> d2 in [0,16), b1 -> d2 in [16,32)
            b0.x = c2[(r0    ) * TT_D2 + lo     ];
            b0.y = c2[(r0 + 1) * TT_D2 + lo     ];
            b1.x = c2[(r0    ) * TT_D2 + 16 + lo];
            b1.y = c2[(r0 + 1) * TT_D2 + 16 + lo];

            acc00 = __builtin_amdgcn_wmma_f32_16x16x4_f32(false, a0, false, b0, (short)0, acc00, false, false);
            acc01 = __builtin_amdgcn_wmma_f32_16x16x4_f32(false, a0, false, b1, (short)0, acc01, false, false);
            acc10 = __builtin_amdgcn_wmma_f32_16x16x4_f32(false, a1, false, b0, (short)0, acc10, false, false);
            acc11 = __builtin_amdgcn_wmma_f32_16x16x4_f32(false, a1, false, b1, (short)0, acc11, false, false);
        }

        // C/D layout: VGPR v holds (M = v + 8*hi, N = lo) per 16x16 tile.
        // flat offset = d1*32 + d2 with d1 = mt*16 + v + 8*hi, d2 = nt*16 + lo.
        float* o = out + (size_t)t * (TT_D1 * TT_D2) + (size_t)(hi * 8) * TT_D2 + lo;
#pragma unroll
        for (int v = 0; v < 8; ++v) {
            __builtin_nontemporal_store(acc00[v], o + (v     ) * TT_D2     );
            __builtin_nontemporal_store(acc01[v], o + (v     ) * TT_D2 + 16);
            __builtin_nontemporal_store(acc10[v], o + (v + 16) * TT_D2     );
            __builtin_nontemporal_store(acc11[v], o + (v + 16) * TT_D2 + 16);
        }
    }
}

extern "C" void kernel_launch(void* const* d_in, const int* in_sizes, int n_in,
                              void* d_out, int out_size, void* d_ws, size_t ws_size,
                              hipStream_t stream) {
    const int*   ids   = (const int*)  d_in[0];  // input_ids (8,4096)
    const float* core1 = (const float*)d_in[1];  // (225,16,32)
    const float* core2 = (const float*)d_in[2];  // (224,16,32)
    const float* phase = (const float*)d_in[3];  // (16,)
    float*       out   = (float*)d_out;

    const int ntok = in_sizes[0];                // 32768 tokens
    const int threads = 256;                     // 8 waves/block (wave32)
    int blocks = 2048;                           // 16384 persistent waves, ~2 tokens each
    int max_blocks = (ntok + 7) / 8;
    if (blocks > max_blocks) blocks = max_blocks;
    if (blocks < 1) blocks = 1;

    tt_embed_wmma_kernel<<<blocks, threads, 0, stream>>>(ids, core1, core2, phase, out, ntok);
}